// gLASE_5317169512922
// MI455X (gfx1250) — compile-verified
//
#include <hip/hip_runtime.h>

#define NN 16384
#define DD 128
#define GD_STEPS 3

typedef __attribute__((ext_vector_type(2))) float v2f;
typedef __attribute__((ext_vector_type(8))) float v8f;

// ---------------------------------------------------------------------------
// Zero a region (float4 granularity). Used for agg+tmp (adjacent, 2*N*D).
// ---------------------------------------------------------------------------
__global__ void zero_f4(float4* __restrict__ p, int n4) {
    int i = blockIdx.x * blockDim.x + threadIdx.x;
    if (i < n4) p[i] = make_float4(0.f, 0.f, 0.f, 0.f);
}

// ---------------------------------------------------------------------------
// agg[dst[e]] += x[src[e]]  — one wave32 per edge, one float4 per lane.
// All data L2-resident (x table = 8 MB << 192 MB L2).
// ---------------------------------------------------------------------------
__global__ void scatter_add_rows(const int* __restrict__ src,
                                 const int* __restrict__ dst,
                                 const float* __restrict__ x,
                                 float* __restrict__ agg, int E) {
    int wave = (blockIdx.x * blockDim.x + threadIdx.x) >> 5;
    int lane = threadIdx.x & 31;
    if (wave >= E) return;
    int s = src[wave];
    int d = dst[wave];
    int off = lane << 2;
    const float4 v = *(const float4*)(x + (size_t)s * DD + off);
    float* o = agg + (size_t)d * DD + off;
    atomicAdd(o + 0, v.x);
    atomicAdd(o + 1, v.y);
    atomicAdd(o + 2, v.z);
    atomicAdd(o + 3, v.w);
}

// ---------------------------------------------------------------------------
// out[i,j] = scale_j * sum_k A[i,k]*W[j,k] (+ addend[i,j])
//   scale_j = q[j] if q != null else 1
// One wave per 16x16 output tile; f32 WMMA 16x16x4, K fully unrolled (32 MACs).
// A-frag: lane(l<16) row M=l holds K={k,k+1}; lanes 16..31 hold K={k+2,k+3}.
// B-frag = Wᵀ tile: column N=l&15 is row (n0+l&15) of W — same addressing.
// C/D: VGPR v -> row v (lanes 0-15) / v+8 (lanes 16-31), col = lane&15.
// ---------------------------------------------------------------------------
__global__ void wmma_gemm_bt(const float* __restrict__ A,
                             const float* __restrict__ W,
                             const float* __restrict__ q,
                             const float* __restrict__ addend,
                             float* __restrict__ out) {
    int wave = blockIdx.x * (blockDim.x >> 5) + (threadIdx.x >> 5);
    int lane = threadIdx.x & 31;
    int row_tile = wave >> 3;          // DD/16 == 8 column tiles
    int col_tile = wave & 7;
    int row0 = row_tile << 4;
    int n0   = col_tile << 4;
    int half = lane >> 4;              // 0: K pair {0,1}, 1: K pair {2,3}
    int l15  = lane & 15;

    const float* arow = A + (size_t)(row0 + l15) * DD + 2 * half;
    const float* brow = W + (size_t)(n0 + l15) * DD + 2 * half;

    v8f acc = {};
#pragma unroll
    for (int k = 0; k < DD; k += 4) {
        v2f a = *(const v2f*)(arow + k);
        v2f b = *(const v2f*)(brow + k);
        acc = __builtin_amdgcn_wmma_f32_16x16x4_f32(
            /*neg_a=*/false, a, /*neg_b=*/false, b,
            /*c_mod=*/(short)0, acc, /*reuse_a=*/false, /*reuse_b=*/false);
    }

    int col = n0 + l15;
    float scale = q ? q[col] : 1.0f;
#pragma unroll
    for (int v = 0; v < 8; ++v) {
        int row = row0 + v + 8 * half;
        float val = acc[v] * scale;
        if (addend) val += addend[(size_t)row * DD + col];
        out[(size_t)row * DD + col] = val;
    }
}

// ---------------------------------------------------------------------------
// Per edge e: attn = <x[s2]*q, x[d2]>;  tmp[d2] += attn * x2[s2]
// One wave32 per edge; wave-wide xor-shuffle reduction for the dot product.
// ---------------------------------------------------------------------------
__global__ void edge_attn_scatter(const int* __restrict__ s2,
                                  const int* __restrict__ d2,
                                  const float* __restrict__ x,
                                  const float* __restrict__ x2,
                                  const float* __restrict__ q,
                                  float* __restrict__ tmp, int E) {
    int wave = (blockIdx.x * blockDim.x + threadIdx.x) >> 5;
    int lane = threadIdx.x & 31;
    if (wave >= E) return;
    int s = s2[wave];
    int d = d2[wave];
    int off = lane << 2;

    const float4 xs = *(const float4*)(x + (size_t)s * DD + off);
    const float4 xd = *(const float4*)(x + (size_t)d * DD + off);
    const float4 qv = *(const float4*)(q + off);

    float p = xs.x * qv.x * xd.x + xs.y * qv.y * xd.y +
              xs.z * qv.z * xd.z + xs.w * qv.w * xd.w;
#pragma unroll
    for (int o = 16; o > 0; o >>= 1) p += __shfl_xor(p, o, 32);

    const float4 v2 = *(const float4*)(x2 + (size_t)s * DD + off);
    float* o = tmp + (size_t)d * DD + off;
    atomicAdd(o + 0, p * v2.x);
    atomicAdd(o + 1, p * v2.y);
    atomicAdd(o + 2, p * v2.z);
    atomicAdd(o + 3, p * v2.w);
}

// ---------------------------------------------------------------------------
// out[i,j] = x1[i,j] - tmp[i,j] * q[j]   (float4 vectorized, col = (idx&31)*4)
// ---------------------------------------------------------------------------
__global__ void combine_kernel(const float* __restrict__ x1,
                               const float* __restrict__ tmp,
                               const float* __restrict__ q,
                               float* __restrict__ out, int n4) {
    int i = blockIdx.x * blockDim.x + threadIdx.x;
    if (i >= n4) return;
    int col = (i & ((DD / 4) - 1)) << 2;
    float4 a  = *(const float4*)(x1 + (size_t)i * 4);
    float4 t  = *(const float4*)(tmp + (size_t)i * 4);
    float4 qv = *(const float4*)(q + col);
    float4 r;
    r.x = a.x - t.x * qv.x;
    r.y = a.y - t.y * qv.y;
    r.z = a.z - t.z * qv.z;
    r.w = a.w - t.w * qv.w;
    *(float4*)(out + (size_t)i * 4) = r;
}

// ---------------------------------------------------------------------------
extern "C" void kernel_launch(void* const* d_in, const int* in_sizes, int n_in,
                              void* d_out, int out_size, void* d_ws, size_t ws_size,
                              hipStream_t stream) {
    const float* x_in = (const float*)d_in[0];
    const int*   ei   = (const int*)d_in[1];   // edge_index  [2, E]
    const int*   ei2  = (const int*)d_in[2];   // edge_index_2 [2, E]
    const float* W1   = (const float*)d_in[3];
    const float* W2   = (const float*)d_in[4];
    const float* Q    = (const float*)d_in[5]; // [1, D] -> D floats

    const int E = in_sizes[1] / 2;
    const size_t ND = (size_t)NN * DD;

    float* ws  = (float*)d_ws;        // needs 4*N*D*4 = 32 MB
    float* agg = ws;                  // [N,D]  zeroed each step
    float* tmp = ws + ND;             // [N,D]  zeroed each step (adjacent)
    float* x1  = ws + 2 * ND;         // [N,D]
    float* x2  = ws + 3 * ND;         // [N,D]
    float* out = (float*)d_out;

    const int edge_blocks = (E * 32 + 255) / 256;        // 1 wave per edge
    const int gemm_blocks = (NN / 16) * (DD / 16) / 8;   // 8 waves/block, 1 tile/wave
    const int zero_blocks = (int)((2 * ND / 4 + 255) / 256);
    const int comb_blocks = (int)((ND / 4 + 255) / 256);

    for (int step = 0; step < GD_STEPS; ++step) {
        const float* x = (step == 0) ? x_in : out;

        // zero agg and tmp in one pass (adjacent in ws)
        zero_f4<<<zero_blocks, 256, 0, stream>>>((float4*)agg, (int)(2 * ND / 4));

        // agg = segment_sum(x[src], dst)
        scatter_add_rows<<<edge_blocks, 256, 0, stream>>>(ei, ei + E, x, agg, E);

        // x1 = x + (agg @ W1^T) * q          (WMMA, scale=q, addend=x)
        wmma_gemm_bt<<<gemm_blocks, 256, 0, stream>>>(agg, W1, Q, x, x1);

        // x2 = x @ W2^T                      (WMMA, no scale, no addend)
        wmma_gemm_bt<<<gemm_blocks, 256, 0, stream>>>(x, W2, nullptr, nullptr, x2);

        // tmp[d2] += <x[s2]*q, x[d2]> * x2[s2]
        edge_attn_scatter<<<edge_blocks, 256, 0, stream>>>(ei2, ei2 + E, x, x2, Q, tmp, E);

        // out = x1 - tmp * q   (out feeds next step as x)
        combine_kernel<<<comb_blocks, 256, 0, stream>>>(x1, tmp, Q, out, (int)(ND / 4));
    }
}